// SiameseGNN_47880295416569
// MI455X (gfx1250) — compile-verified
//
#include <hip/hip_runtime.h>
#include <hip/hip_bf16.h>

// ---------------------------------------------------------------------------
// Siamese 5-layer GCN + MLP for MI455X (gfx1250, wave32).
//  - Dense per-node GEMMs: V_WMMA_F32_16X16X4_F32 (fp32 matrix path).
//  - Weight tile (64x64 f32, 16KB) staged into LDS per block via the
//    Tensor Data Mover (tensor_load_to_lds + s_wait_tensorcnt).
//  - Edge segment-sums: relaxed agent-scope float atomics (L2-resident).
// ---------------------------------------------------------------------------

typedef float v8f __attribute__((ext_vector_type(8)));
typedef float v2f __attribute__((ext_vector_type(2)));
typedef unsigned int u32x4 __attribute__((ext_vector_type(4)));
typedef int i32x8 __attribute__((ext_vector_type(8)));
typedef int i32x4 __attribute__((ext_vector_type(4)));

#define NND   50000
#define NE    1200000
#define FF    64
#define GG    128
#define FCC   256
#define EMBB  128

static __device__ __forceinline__ void atomAddF(float* p, float v) {
    __hip_atomic_fetch_add(p, v, __ATOMIC_RELAXED, __HIP_MEMORY_SCOPE_AGENT);
}

// -------------------------------- utility ----------------------------------
__global__ void fill0_kernel(float* __restrict__ p, int n) {
    int i = blockIdx.x * blockDim.x + threadIdx.x;
    if (i < n) p[i] = 0.0f;
}

// ------------------------------ degree path --------------------------------
__global__ void deg_scatter_kernel(const int* __restrict__ dst, float* __restrict__ deg, int ne) {
    int e = blockIdx.x * blockDim.x + threadIdx.x;
    if (e < ne) atomAddF(&deg[dst[e]], 1.0f);
}

__global__ void deg_to_dinv_kernel(float* __restrict__ d, int n) {
    int i = blockIdx.x * blockDim.x + threadIdx.x;
    if (i < n) d[i] = rsqrtf(d[i] + 1.0f);   // self-loop included in degree
}

// ------------------------- WMMA fp32 GEMM: h = x @ W -----------------------
// Grid.x = N/16 row tiles; 128 threads = 4 waves; wave w owns cols [16w,16w+16).
// W (64x64 f32) is DMA'd into LDS once per block by the Tensor Data Mover.
// K = 64 swept with 16 chained V_WMMA_F32_16X16X4_F32.
__global__ void gcn_gemm_wmma_kernel(const float* __restrict__ x,
                                     const float* __restrict__ W,   // [64,64] row-major (k,n)
                                     float* __restrict__ h, int nrows) {
    __shared__ float Wlds[FF * FF];

    const int lane = threadIdx.x & 31;
    const int wave = threadIdx.x >> 5;

    if (wave == 0) {
        // --- Tensor DMA descriptor: 1 row x 4096 f32 elements, global -> LDS ---
        const unsigned ldsOff = (unsigned)(uintptr_t)Wlds;        // LDS aperture low bits
        const unsigned long long ga = (unsigned long long)(uintptr_t)W;
        u32x4 g0;
        g0[0] = 1u;                                               // count=1 (valid D#)
        g0[1] = ldsOff;                                           // lds_addr (bytes)
        g0[2] = (unsigned)(ga & 0xFFFFFFFFu);                     // global_addr[31:0]
        g0[3] = ((unsigned)(ga >> 32) & 0x01FFFFFFu) | (2u << 30);// addr[56:32] | type=2
        i32x8 g1 = {};
        g1[0] = (int)(2u << 16);        // workgroup_mask=0 | data_size=2 (4 bytes)
        g1[1] = (int)(4096u << 16);     // tensor_dim0[15:0]=4096 in bits[31:16]
        g1[2] = (int)(1u << 16);        // tensor_dim0[31:16]=0 | tensor_dim1[15:0]=1
        g1[3] = (int)(4096u << 16);     // tensor_dim1[31:16]=0 | tile_dim0=4096
        g1[4] = 0;                      // tile_dim1=0 (unused) | tile_dim2=0
        g1[5] = 4096;                   // tensor_dim0_stride[31:0]
        g1[6] = 0;                      // stride hi | tensor_dim1_stride lo
        g1[7] = 0;
        i32x4 gz = {};
        i32x8 gz8 = {};
        __builtin_amdgcn_tensor_load_to_lds(g0, g1, gz, gz, gz8, 0);
        __builtin_amdgcn_s_wait_tensorcnt(0);                     // TENSORcnt -> 0
    }
    __syncthreads();                                              // LDS W visible to all waves

    const int m0  = blockIdx.x * 16;
    const int n0  = wave * 16;
    const int sel = lane & 15;           // row (A) / col (B) within tile
    const int hi  = lane >> 4;           // 0: K pair {0,1}; 1: K pair {2,3}

    int arow = m0 + sel;
    if (arow >= nrows) arow = nrows - 1;             // N%16==0 in practice
    const float* __restrict__ xrow = x + (size_t)arow * FF;

    v8f c = {};
#pragma unroll
    for (int k0 = 0; k0 < FF; k0 += 4) {
        const int kb = k0 + 2 * hi;
        v2f a, b;
        a.x = xrow[kb];
        a.y = xrow[kb + 1];
        b.x = Wlds[kb * FF + n0 + sel];
        b.y = Wlds[(kb + 1) * FF + n0 + sel];
        // 8 args: (neg_a, A, neg_b, B, c_mod, C, reuse_a, reuse_b)
        c = __builtin_amdgcn_wmma_f32_16x16x4_f32(false, a, false, b,
                                                  (short)0, c, false, false);
    }
#pragma unroll
    for (int r = 0; r < 8; ++r) {
        const int row = m0 + r + 8 * hi;             // C/D layout: VGPR r, lane half
        if (row < nrows) h[(size_t)row * FF + n0 + sel] = c[r];
    }
}

// --------------------- edge gather/scatter (segment sum) -------------------
// One thread per (edge, 4-float chunk): 16 chunks cover 64 features.
__global__ void edge_scatter_kernel(const int* __restrict__ src,
                                    const int* __restrict__ dst,
                                    const float* __restrict__ dinv,
                                    const float* __restrict__ h,
                                    float* __restrict__ agg, int ne) {
    long long tid = (long long)blockIdx.x * blockDim.x + threadIdx.x;
    if (tid >= (long long)ne * 16) return;
    const int e = (int)(tid >> 4);
    const int c = (int)(tid & 15) * 4;
    const int s = src[e], d = dst[e];
    const float norm = dinv[s] * dinv[d];
    const float4 hv = *(const float4*)(h + (size_t)s * FF + c);
    float* ap = agg + (size_t)d * FF + c;
    atomAddF(ap + 0, hv.x * norm);
    atomAddF(ap + 1, hv.y * norm);
    atomAddF(ap + 2, hv.z * norm);
    atomAddF(ap + 3, hv.w * norm);
}

// ------------------ x' = agg + h*dinv^2 + b  (+ ReLU) ----------------------
__global__ void finalize_kernel(const float* __restrict__ agg,
                                const float* __restrict__ h,
                                const float* __restrict__ dinv,
                                const float* __restrict__ b,
                                float* __restrict__ xout, int n, int relu) {
    int i = blockIdx.x * blockDim.x + threadIdx.x;
    if (i >= n) return;
    const int node = i >> 6;
    const int col  = i & 63;
    const float di = dinv[node];
    float v = agg[i] + h[i] * di * di + b[col];
    if (relu) v = fmaxf(v, 0.0f);
    xout[i] = v;
}

// ------------------------------ mean pool ----------------------------------
__global__ void pool_scatter_kernel(const float* __restrict__ x,
                                    const int* __restrict__ batch,
                                    float* __restrict__ pooled,
                                    float* __restrict__ counts, int n) {
    int tid = blockIdx.x * blockDim.x + threadIdx.x;
    if (tid >= n * 16) return;
    const int node = tid >> 4;
    const int c    = (tid & 15) * 4;
    const int g    = batch[node];
    const float4 v = *(const float4*)(x + (size_t)node * FF + c);
    float* pp = pooled + (size_t)g * FF + c;
    atomAddF(pp + 0, v.x);
    atomAddF(pp + 1, v.y);
    atomAddF(pp + 2, v.z);
    atomAddF(pp + 3, v.w);
    if (c == 0) atomAddF(&counts[g], 1.0f);
}

// -------------------------------- MLP --------------------------------------
__global__ void mlp1_kernel(const float* __restrict__ pooled,
                            const float* __restrict__ counts,
                            const float* __restrict__ w,   // [64,256]
                            const float* __restrict__ b,
                            float* __restrict__ out) {     // [G,256]
    const int g = blockIdx.x, t = threadIdx.x;
    const float inv = 1.0f / fmaxf(counts[g], 1.0f);
    float acc = b[t];
    const float* pr = pooled + (size_t)g * FF;
#pragma unroll 8
    for (int k = 0; k < FF; ++k) acc += pr[k] * inv * w[(size_t)k * FCC + t];
    out[(size_t)g * FCC + t] = fmaxf(acc, 0.0f);
}

__global__ void mlp2_kernel(const float* __restrict__ hin,  // [G,256]
                            const float* __restrict__ w,    // [256,256]
                            const float* __restrict__ b,
                            float* __restrict__ out) {      // [G,256]
    const int g = blockIdx.x, t = threadIdx.x;
    float acc = b[t];
    const float* hr = hin + (size_t)g * FCC;
#pragma unroll 8
    for (int k = 0; k < FCC; ++k) acc += hr[k] * w[(size_t)k * FCC + t];
    out[(size_t)g * FCC + t] = fmaxf(acc, 0.0f);
}

__global__ void mlp3_kernel(const float* __restrict__ hin,  // [G,256]
                            const float* __restrict__ w,    // [256,128]
                            const float* __restrict__ b,
                            float* __restrict__ out) {      // [G,128]
    const int g = blockIdx.x, t = threadIdx.x;
    float acc = b[t];
    const float* hr = hin + (size_t)g * FCC;
#pragma unroll 8
    for (int k = 0; k < FCC; ++k) acc += hr[k] * w[(size_t)k * EMBB + t];
    out[(size_t)g * EMBB + t] = acc;
}

// ------------------------------ launcher -----------------------------------
static inline int cdiv(long long a, int b) { return (int)((a + b - 1) / b); }

extern "C" void kernel_launch(void* const* d_in, const int* in_sizes, int n_in,
                              void* d_out, int out_size, void* d_ws, size_t ws_size,
                              hipStream_t stream) {
    (void)in_sizes; (void)n_in; (void)out_size; (void)ws_size;

    const float* convW = (const float*)d_in[9];
    const float* convb = (const float*)d_in[10];
    const float* fw1 = (const float*)d_in[11];
    const float* fb1 = (const float*)d_in[12];
    const float* fw2 = (const float*)d_in[13];
    const float* fb2 = (const float*)d_in[14];
    const float* fw3 = (const float*)d_in[15];
    const float* fb3 = (const float*)d_in[16];

    float* ws = (float*)d_ws;
    float* bufX   = ws;                 ws += (size_t)NND * FF;
    float* bufH   = ws;                 ws += (size_t)NND * FF;
    float* bufAgg = ws;                 ws += (size_t)NND * FF;
    float* dinvb  = ws;                 ws += NND;
    float* pooled = ws;                 ws += GG * FF;
    float* counts = ws;                 ws += GG;
    float* h1     = ws;                 ws += GG * FCC;
    float* h2     = ws;                 ws += GG * FCC;

    for (int br = 0; br < 3; ++br) {
        const float* xin  = (const float*)d_in[br * 3 + 0];
        const int*   eidx = (const int*)  d_in[br * 3 + 1];
        const int*   bat  = (const int*)  d_in[br * 3 + 2];
        const int* srcp = eidx;          // edge_index[0]
        const int* dstp = eidx + NE;     // edge_index[1]

        // degrees -> dinv
        fill0_kernel<<<cdiv(NND, 256), 256, 0, stream>>>(dinvb, NND);
        deg_scatter_kernel<<<cdiv(NE, 256), 256, 0, stream>>>(dstp, dinvb, NE);
        deg_to_dinv_kernel<<<cdiv(NND, 256), 256, 0, stream>>>(dinvb, NND);

        const float* xcur = xin;
        for (int l = 0; l < 5; ++l) {
            const float* Wl = convW + (size_t)l * FF * FF;
            const float* bl = convb + (size_t)l * FF;
            gcn_gemm_wmma_kernel<<<cdiv(NND, 16), 128, 0, stream>>>(xcur, Wl, bufH, NND);
            fill0_kernel<<<cdiv((long long)NND * FF, 256), 256, 0, stream>>>(bufAgg, NND * FF);
            edge_scatter_kernel<<<cdiv((long long)NE * 16, 256), 256, 0, stream>>>(
                srcp, dstp, dinvb, bufH, bufAgg, NE);
            finalize_kernel<<<cdiv((long long)NND * FF, 256), 256, 0, stream>>>(
                bufAgg, bufH, dinvb, bl, bufX, NND * FF, (l < 4) ? 1 : 0);
            xcur = bufX;
        }

        // global mean pool
        fill0_kernel<<<cdiv(GG * FF, 256), 256, 0, stream>>>(pooled, GG * FF);
        fill0_kernel<<<1, 256, 0, stream>>>(counts, GG);
        pool_scatter_kernel<<<cdiv((long long)NND * 16, 256), 256, 0, stream>>>(
            xcur, bat, pooled, counts, NND);

        // MLP head
        mlp1_kernel<<<GG, FCC, 0, stream>>>(pooled, counts, fw1, fb1, h1);
        mlp2_kernel<<<GG, FCC, 0, stream>>>(h1, fw2, fb2, h2);
        mlp3_kernel<<<GG, EMBB, 0, stream>>>(h2, fw3, fb3,
                                             (float*)d_out + (size_t)br * GG * EMBB);
    }
}